// Protocl_76476187673023
// MI455X (gfx1250) — compile-verified
//
#include <hip/hip_runtime.h>
#include <math.h>

// loss = (1/N) * sum_n sqrt( sum_d (x[n,d] - proto[label_n,d] + EPS)^2 )
// The reference's 1/neg_dists.sum() term is exactly 1/inf == 0.0 (each row of
// neg_dists contains exactly one +inf), so the O(N*K*D) GEMM contributes
// nothing and the kernel is a pure gather + row-reduction: ~102 MiB of traffic
// -> ~4.5 us at 23.3 TB/s HBM. The per-row sum over D runs on the matrix pipe
// via V_WMMA_F32_16X16X4_F32 with an all-ones B matrix (D[m,n] += sum_k A[m,k]),
// using two independent accumulator chains so back-to-back WMMAs have no
// D->C RAW hazard (ISA 7.12.1).

typedef float v2f __attribute__((ext_vector_type(2)));
typedef float v8f __attribute__((ext_vector_type(8)));

#define D_DIM 768
#define EPS_F 1e-6f

__global__ void Protocl_zero_kernel(float* out) { *out = 0.0f; }

__global__ __launch_bounds__(256) void Protocl_pos_loss_kernel(
    const float* __restrict__ x, const float* __restrict__ protos,
    const int* __restrict__ labels, float* __restrict__ out, int n) {
  const int lane = threadIdx.x & 31;      // wave32
  const int wave = threadIdx.x >> 5;      // 8 waves / block
  const int half = lane >> 4;             // A-matrix K-half: 0 -> K=0,1 ; 1 -> K=2,3
  const int m    = lane & 15;             // A-matrix row within the 16-row tile
  const int row  = blockIdx.x * 128 + wave * 16 + m;   // N % 128 == 0

  const long long lab = (long long)labels[row];
  const float* __restrict__ xr = x      + (long long)row * D_DIM + (half << 1);
  const float* __restrict__ pr = protos + lab            * D_DIM + (half << 1);

  v8f acc0 = {0.f, 0.f, 0.f, 0.f, 0.f, 0.f, 0.f, 0.f};
  v8f acc1 = {0.f, 0.f, 0.f, 0.f, 0.f, 0.f, 0.f, 0.f};
  const v2f ones = {1.0f, 1.0f};  // B matrix = all ones -> row-sum reduction

  // 96 iterations, 2 independent WMMA chains (even 4-col chunk -> acc0,
  // odd chunk -> acc1). x is streamed non-temporally (no reuse; keep L2 for
  // the 6 MiB prototype table which is reused ~16x).
  #pragma unroll 2
  for (int d = 0; d < D_DIM; d += 8) {
    v2f xv0 = __builtin_nontemporal_load((const v2f*)(xr + d));
    v2f pv0 = *(const v2f*)(pr + d);
    v2f xv1 = __builtin_nontemporal_load((const v2f*)(xr + d + 4));
    v2f pv1 = *(const v2f*)(pr + d + 4);

    v2f df0 = (xv0 - pv0) + EPS_F;
    v2f a0  = df0 * df0;
    v2f df1 = (xv1 - pv1) + EPS_F;
    v2f a1  = df1 * df1;

    // (neg_a, A, neg_b, B, c_mod, C, reuse_a, reuse_b)
    acc0 = __builtin_amdgcn_wmma_f32_16x16x4_f32(false, a0, false, ones,
                                                 (short)0, acc0, false, false);
    acc1 = __builtin_amdgcn_wmma_f32_16x16x4_f32(false, a1, false, ones,
                                                 (short)0, acc1, false, false);
  }

  v8f acc = acc0 + acc1;

  // 16x16 f32 C/D layout: lane 0,  VGPR v -> D[v, 0]   = rowsum(base + v)
  //                       lane 16, VGPR v -> D[8+v, 0] = rowsum(base + 8 + v)
  if (m == 0) {
    float s = 0.0f;
    #pragma unroll
    for (int v = 0; v < 8; ++v) s += __builtin_amdgcn_sqrtf(acc[v]);
    atomicAdd(out, s * (1.0f / (float)n));
  }
}

extern "C" void kernel_launch(void* const* d_in, const int* in_sizes, int n_in,
                              void* d_out, int out_size, void* d_ws, size_t ws_size,
                              hipStream_t stream) {
  const float* x      = (const float*)d_in[0];   // [N, 768] fp32
  const float* protos = (const float*)d_in[1];   // [K, 768] fp32
  const int*   labels = (const int*)d_in[2];     // [N] int32 (JAX downcasts int64)
  float* out = (float*)d_out;                    // scalar fp32 loss
  const int n = in_sizes[2];                     // N = 32768

  Protocl_zero_kernel<<<1, 1, 0, stream>>>(out);
  Protocl_pos_loss_kernel<<<n / 128, 256, 0, stream>>>(x, protos, labels, out, n);
}